// HashEncoder_87943750353146
// MI455X (gfx1250) — compile-verified
//
#include <hip/hip_runtime.h>
#include <hip/hip_bf16.h>
#include <stdint.h>

// Instant-NGP style multiresolution hash-grid encoder for MI455X (gfx1250).
// 16 levels, 2 features/level, 19-bit hashmap, trilinear interpolation.
// Gather-bound: 64 MB of tables fit entirely in the 192 MB L2; x/out are
// streamed with non-temporal hints so the tables stay L2-resident.

typedef float v4f __attribute__((ext_vector_type(4)));  // native vector for NT store
typedef float v2f __attribute__((ext_vector_type(2)));

constexpr int      kLevels   = 16;
constexpr uint32_t kHashMask = (1u << 19) - 1u;   // HASHMAP_SIZE - 1
constexpr uint32_t kHashSize = 1u << 19;

__device__ __forceinline__ v2f level_encode(const v2f* __restrict__ tab,
                                            float res,
                                            float px, float py, float pz) {
  // scale + floor(trunc) + frac, matching x.astype(int32) semantics (x >= 0)
  float sx = px * res, sy = py * res, sz = pz * res;
  int ix = (int)sx, iy = (int)sy, iz = (int)sz;
  float fx = sx - (float)ix;
  float fy = sy - (float)iy;
  float fz = sz - (float)iz;

  // hash partials; PI = {1, 2654435761, 805459861}; (k+1)*P == k*P + P mod 2^32
  uint32_t hx0 = (uint32_t)ix,                 hx1 = hx0 + 1u;
  uint32_t hy0 = (uint32_t)iy * 2654435761u,   hy1 = hy0 + 2654435761u;
  uint32_t hz0 = (uint32_t)iz * 805459861u,    hz1 = hz0 + 805459861u;

  uint32_t i0 = (hx0 ^ hy0 ^ hz0) & kHashMask; // (0,0,0)
  uint32_t i1 = (hx1 ^ hy0 ^ hz0) & kHashMask; // (1,0,0)
  uint32_t i2 = (hx0 ^ hy1 ^ hz0) & kHashMask; // (0,1,0)
  uint32_t i3 = (hx1 ^ hy1 ^ hz0) & kHashMask; // (1,1,0)
  uint32_t i4 = (hx0 ^ hy0 ^ hz1) & kHashMask; // (0,0,1)
  uint32_t i5 = (hx1 ^ hy0 ^ hz1) & kHashMask; // (1,0,1)
  uint32_t i6 = (hx0 ^ hy1 ^ hz1) & kHashMask; // (0,1,1)
  uint32_t i7 = (hx1 ^ hy1 ^ hz1) & kHashMask; // (1,1,1)

  // issue all 8 gathers before any use -> compiler can clause the b64 loads
  v2f f0 = tab[i0];
  v2f f1 = tab[i1];
  v2f f2 = tab[i2];
  v2f f3 = tab[i3];
  v2f f4 = tab[i4];
  v2f f5 = tab[i5];
  v2f f6 = tab[i6];
  v2f f7 = tab[i7];

  // trilinear weights (overlaps with load latency)
  float gx = 1.0f - fx, gy = 1.0f - fy, gz = 1.0f - fz;
  float w00 = gy * gz, w10 = fy * gz, w01 = gy * fz, w11 = fy * fz;
  float w0 = gx * w00, w1 = fx * w00;
  float w2 = gx * w10, w3 = fx * w10;
  float w4 = gx * w01, w5 = fx * w01;
  float w6 = gx * w11, w7 = fx * w11;

  float o0 = w0 * f0.x;
  float o1 = w0 * f0.y;
  o0 = fmaf(w1, f1.x, o0);  o1 = fmaf(w1, f1.y, o1);
  o0 = fmaf(w2, f2.x, o0);  o1 = fmaf(w2, f2.y, o1);
  o0 = fmaf(w3, f3.x, o0);  o1 = fmaf(w3, f3.y, o1);
  o0 = fmaf(w4, f4.x, o0);  o1 = fmaf(w4, f4.y, o1);
  o0 = fmaf(w5, f5.x, o0);  o1 = fmaf(w5, f5.y, o1);
  o0 = fmaf(w6, f6.x, o0);  o1 = fmaf(w6, f6.y, o1);
  o0 = fmaf(w7, f7.x, o0);  o1 = fmaf(w7, f7.y, o1);
  v2f r; r.x = o0; r.y = o1;
  return r;
}

__global__ __launch_bounds__(256) void hashgrid_encode_kernel(
    const float* __restrict__ x,        // (N,3)
    const float* __restrict__ tables,   // (16, 2^19, 2)
    float* __restrict__ out,            // (N, 32) level-major
    int n) {
  int tid = blockIdx.x * blockDim.x + threadIdx.x;
  if (tid >= n) return;

  // resolutions: int(16 * (32^(1/15))^i), i=0..15
  constexpr float kRes[kLevels] = {
      16.f, 20.f, 25.f, 32.f, 40.f, 50.f, 64.f, 80.f,
      101.f, 128.f, 161.f, 203.f, 256.f, 322.f, 406.f, 512.f};

  // stream coords non-temporally (read-once; don't pollute L2 table residency)
  const float* xp = x + (size_t)tid * 3;
  float px = __builtin_nontemporal_load(xp + 0);
  float py = __builtin_nontemporal_load(xp + 1);
  float pz = __builtin_nontemporal_load(xp + 2);

  v4f* __restrict__ outv = (v4f*)(out + (size_t)tid * (kLevels * 2));

#pragma unroll
  for (int l = 0; l < kLevels; l += 2) {
    const v2f* t0 = (const v2f*)tables + ((size_t)l << 19);
    const v2f* t1 = t0 + kHashSize;
    v2f a = level_encode(t0, kRes[l],     px, py, pz);
    v2f b = level_encode(t1, kRes[l + 1], px, py, pz);
    // two adjacent levels are contiguous in the (N, L*F) output: one b128 NT store
    v4f v; v.x = a.x; v.y = a.y; v.z = b.x; v.w = b.y;
    __builtin_nontemporal_store(v, outv + (l >> 1));
  }
}

extern "C" void kernel_launch(void* const* d_in, const int* in_sizes, int n_in,
                              void* d_out, int out_size, void* d_ws, size_t ws_size,
                              hipStream_t stream) {
  const float* x      = (const float*)d_in[0];   // (N,3) float32
  const float* tables = (const float*)d_in[1];   // (16, 2^19, 2) float32
  float* out          = (float*)d_out;           // (N, 32) float32

  int n = in_sizes[0] / 3;
  const int block = 256;                         // 8 wave32 waves per workgroup
  const int grid  = (n + block - 1) / block;
  hipLaunchKernelGGL(hashgrid_encode_kernel, dim3(grid), dim3(block), 0, stream,
                     x, tables, out, n);
}